// Attention_Net_68685116997830
// MI455X (gfx1250) — compile-verified
//
#include <hip/hip_runtime.h>
#include <hip/hip_bf16.h>
#include <math.h>

// ---------------------------------------------------------------------------
// NTS-Net proposal head for MI455X (gfx1250, wave32, WMMA).
//   conv1 (2048->128, 3x3) dominates at ~7.4 GFLOP -> fp32 WMMA 16x16x4
//   (fp32 keeps NMS argmax bit-faithful to the f32 reference).
//   Total traffic ~50MB << 192MB L2 -> compute/issue bound, so the inner
//   loop is tuned to minimize loads per WMMA: A via LDS b64, B via packed b64.
// ---------------------------------------------------------------------------

typedef __attribute__((ext_vector_type(2))) float v2f;
typedef __attribute__((ext_vector_type(8))) float v8f;

#define IMG  448
#define PAD  224
#define OUTS 224
#define NANCH 2793
#define BATCH 8
#define KC   64           // K-chunk (channels) staged in LDS per step
#define KCP  68           // padded LDS row stride (floats) to avoid bank conflicts

// ---------------------------------------------------------------------------
// Weight repack: OIHW w[o][c][r] -> wt[r][o][c]  (K contiguous for b64 loads)
// ---------------------------------------------------------------------------
__global__ void repack_w(const float* __restrict__ w, float* __restrict__ wt, int Cin)
{
    int gid = blockIdx.x * 256 + threadIdx.x;
    const int n = 128 * Cin * 9;
    if (gid >= n) return;
    const int c = gid % Cin;
    const int o = (gid / Cin) % 128;
    const int r = gid / (Cin * 128);
    wt[gid] = w[((size_t)o * Cin + c) * 9 + r];
}

// ---------------------------------------------------------------------------
// 3x3 conv + bias + ReLU as implicit GEMM with V_WMMA_F32_16X16X4_F32.
// Block = 256 threads = 8 waves; block owns a 16(M) x 128(N) output tile,
// wave w owns N-tile [16w, 16w+16). K = Cin*9, iterated as 9 shifted 1x1
// GEMMs so both A and (packed) B stay channel-contiguous.
// PACKED=true : B fragment = one aligned global_load_b64 from wt[r][o][c].
// PACKED=false: fallback strided loads from original OIHW weights.
// ---------------------------------------------------------------------------
template <bool PACKED>
__global__ __launch_bounds__(256)
void conv3x3_relu_wmma(const float* __restrict__ in, const float* __restrict__ w,
                       const float* __restrict__ bias, float* __restrict__ out,
                       int Cin, int Hin, int Hout, int stride)
{
    const int b     = blockIdx.y;
    const int mBase = blockIdx.x * 16;
    const int M     = Hout * Hout;
    const int lane  = threadIdx.x & 31;
    const int wv    = threadIdx.x >> 5;   // 0..7 : N-tile index
    const int hlf   = lane >> 4;          // lane half (K split per ISA layout)
    const int lm    = lane & 15;          // A row / B,C,D column within tile
    const int o     = wv * 16 + lm;       // output channel handled by this lane

    __shared__ float Als[16 * KCP];

    v8f acc = {};

    const size_t inB = (size_t)b * Cin * Hin * Hin;
    const size_t HW  = (size_t)Hin * Hin;

    for (int r = 0; r < 9; ++r) {
        const int ky = r / 3, kx = r % 3;
        for (int cBase = 0; cBase < Cin; cBase += KC) {
            // ---- stage A[16 rows][KC channels] into LDS (zero-padded edges) ----
            for (int e = (int)threadIdx.x; e < 16 * KC; e += 256) {
                const int m = e & 15;
                const int c = e >> 4;
                const int s = mBase + m;
                float v = 0.f;
                if (s < M) {
                    const int y  = s / Hout, x = s % Hout;
                    const int iy = y * stride + ky - 1;
                    const int ix = x * stride + kx - 1;
                    if (iy >= 0 && iy < Hin && ix >= 0 && ix < Hin)
                        v = in[inB + (size_t)(cBase + c) * HW + (size_t)iy * Hin + ix];
                }
                Als[m * KCP + c] = v;
            }
            // prefetch next channel chunk into the hierarchy (global_prefetch_b8)
            if (cBase + KC < Cin)
                __builtin_prefetch(in + inB + (size_t)(cBase + KC + (threadIdx.x >> 4)) * HW, 0, 1);
            __syncthreads();

            // ---- 16 WMMA steps over this K-chunk ----
            // A fragment (16x4 f32): lanes 0-15 hold K=k,k+1 ; lanes 16-31 hold K=k+2,k+3
            // B fragment (4x16 f32): mirrored K split, column = lm.
            const float* wp = PACKED
                ? (w + ((size_t)r * 128 + o) * Cin + cBase)
                : (w + (size_t)o * Cin * 9 + (size_t)cBase * 9 + r);
            #pragma unroll 8
            for (int cc = 0; cc < KC; cc += 4) {
                const int k2 = cc + 2 * hlf;
                v2f a = *(const v2f*)&Als[lm * KCP + k2];
                v2f bb;
                if (PACKED) {
                    bb = *(const v2f*)&wp[k2];            // one global_load_b64
                } else {
                    bb.x = wp[(size_t)k2 * 9];
                    bb.y = wp[(size_t)(k2 + 1) * 9];
                }
                acc = __builtin_amdgcn_wmma_f32_16x16x4_f32(
                        /*neg_a=*/false, a, /*neg_b=*/false, bb,
                        /*c_mod=*/(short)0, acc, /*reuse_a=*/false, /*reuse_b=*/false);
            }
            __syncthreads();
        }
    }

    // ---- epilogue: bias + ReLU, C/D layout: VGPR v -> row m = v + 8*half ----
    const float bv = bias[o];
    #pragma unroll
    for (int v = 0; v < 8; ++v) {
        const int s = mBase + v + 8 * hlf;
        if (s < M)
            out[((size_t)b * 128 + o) * M + s] = fmaxf(acc[v] + bv, 0.f);
    }
}

// ---------------------------------------------------------------------------
// 1x1 score heads: scores[b][a] for a in [0,2793). Tiny (5.7 MFLOP).
// ---------------------------------------------------------------------------
__global__ void scores_kernel(const float* __restrict__ d1, const float* __restrict__ d2,
                              const float* __restrict__ d3,
                              const float* __restrict__ tw1, const float* __restrict__ tb1,
                              const float* __restrict__ tw2, const float* __restrict__ tb2,
                              const float* __restrict__ tw3, const float* __restrict__ tb3,
                              float* __restrict__ scores)
{
    int gid = blockIdx.x * blockDim.x + threadIdx.x;
    if (gid >= BATCH * NANCH) return;
    int b = gid / NANCH;
    int a = gid % NANCH;
    const float *feat, *tw, *tb;
    int HW, ch, s;
    if (a < 1176)      { feat = d1; tw = tw1; tb = tb1; HW = 196; ch = a / 196;        s = a % 196; }
    else if (a < 2352) { feat = d2; tw = tw2; tb = tb2; HW = 196; ch = (a-1176) / 196; s = (a-1176) % 196; }
    else               { feat = d3; tw = tw3; tb = tb3; HW = 49;  ch = (a-2352) / 49;  s = (a-2352) % 49; }
    const float* f  = feat + (size_t)b * 128 * HW + s;
    const float* wv = tw + ch * 128;
    float acc = tb[ch];
    #pragma unroll 8
    for (int c = 0; c < 128; ++c) acc = fmaf(f[(size_t)c * HW], wv[c], acc);
    scores[gid] = acc;
}

// ---------------------------------------------------------------------------
// NMS top-4 per batch. One block per image; scores live in LDS.
// Argmax tie-break = lowest index (matches jnp.argmax). Suppress iou>=0.5.
// ---------------------------------------------------------------------------
__global__ __launch_bounds__(256)
void nms_kernel(const float* __restrict__ scores, const int* __restrict__ ea,
                int* __restrict__ topIdx, float* __restrict__ outProbIdx)
{
    const int b   = blockIdx.x;
    const int tid = threadIdx.x;
    __shared__ float ssc[NANCH];
    __shared__ float rv[256];
    __shared__ int   ri[256];
    __shared__ int   sel;

    const float* sb = scores + b * NANCH;
    for (int i = tid; i < NANCH; i += 256) ssc[i] = sb[i];
    __syncthreads();

    for (int t = 0; t < 4; ++t) {
        float v = -__builtin_inff(); int idx = NANCH;
        for (int i = tid; i < NANCH; i += 256)
            if (ssc[i] > v) { v = ssc[i]; idx = i; }    // increasing i keeps first max
        rv[tid] = v; ri[tid] = idx;
        __syncthreads();
        for (int off = 128; off > 0; off >>= 1) {
            if (tid < off) {
                float v2 = rv[tid + off]; int i2 = ri[tid + off];
                if (v2 > rv[tid] || (v2 == rv[tid] && i2 < ri[tid])) { rv[tid] = v2; ri[tid] = i2; }
            }
            __syncthreads();
        }
        if (tid == 0) {
            sel = ri[0];
            topIdx[b * 4 + t]           = sel;
            outProbIdx[b * 4 + t]       = sb[sel];       // prob from ORIGINAL scores
            outProbIdx[32 + b * 4 + t]  = (float)sel;    // index (flat float output)
        }
        __syncthreads();
        const int si = sel;
        const float y0 = (float)ea[si*4+0], x0 = (float)ea[si*4+1];
        const float y1 = (float)ea[si*4+2], x1 = (float)ea[si*4+3];
        const float a1 = (y1 - y0) * (x1 - x0);
        for (int i = tid; i < NANCH; i += 256) {
            float by0 = (float)ea[i*4+0], bx0 = (float)ea[i*4+1];
            float by1 = (float)ea[i*4+2], bx1 = (float)ea[i*4+3];
            float yy0 = fmaxf(y0, by0), xx0 = fmaxf(x0, bx0);
            float yy1 = fminf(y1, by1), xx1 = fminf(x1, bx1);
            float inter = fmaxf(yy1 - yy0, 0.f) * fmaxf(xx1 - xx0, 0.f);
            float a2 = (by1 - by0) * (bx1 - bx0);
            if (inter / (a1 + a2 - inter) >= 0.5f) ssc[i] = -__builtin_inff();
        }
        __syncthreads();
    }
}

// ---------------------------------------------------------------------------
// Bilinear crop-resize from the implicitly zero-padded 896x896 image.
// One thread per output pixel; pure gather, HBM-bound (~100MB -> ~4us).
// ---------------------------------------------------------------------------
__global__ void crop_kernel(const float* __restrict__ x, const int* __restrict__ ea,
                            const int* __restrict__ topIdx, float* __restrict__ out)
{
    const int n = 32 * 3 * OUTS * OUTS;
    int gid = blockIdx.x * blockDim.x + threadIdx.x;
    if (gid >= n) return;
    const int px = gid % OUTS;
    const int py = (gid / OUTS) % OUTS;
    const int c  = (gid / (OUTS * OUTS)) % 3;
    const int k  = gid / (OUTS * OUTS * 3);
    const int b  = k >> 2;
    const int si = topIdx[k];

    const float y0 = (float)ea[si*4+0], x0 = (float)ea[si*4+1];
    const float y1 = (float)ea[si*4+2], x1 = (float)ea[si*4+3];
    const float ty = (float)py / (OUTS - 1.0f);
    const float tx = (float)px / (OUTS - 1.0f);
    const float ys = y0 + (y1 - y0 - 1.f) * ty;
    const float xs = x0 + (x1 - x0 - 1.f) * tx;
    const float yf = floorf(ys), xf = floorf(xs);
    const float wy = ys - yf,  wx = xs - xf;
    const int hi = IMG + 2 * PAD - 1;             // 895
    int y0i = min(max((int)yf, 0), hi);
    int y1i = min(y0i + 1, hi);
    int x0i = min(max((int)xf, 0), hi);
    int x1i = min(x0i + 1, hi);

    auto samp = [&](int yy, int xx) -> float {
        int iy = yy - PAD, ix = xx - PAD;
        if ((unsigned)iy >= (unsigned)IMG || (unsigned)ix >= (unsigned)IMG) return 0.f;
        return x[(((size_t)b * 3 + c) * IMG + iy) * IMG + ix];
    };
    const float Ia = samp(y0i, x0i), Ib = samp(y0i, x1i);
    const float Ic = samp(y1i, x0i), Id = samp(y1i, x1i);
    const float top = Ia * (1.f - wx) + Ib * wx;
    const float bot = Ic * (1.f - wx) + Id * wx;
    out[gid] = top * (1.f - wy) + bot * wy;
}

// ---------------------------------------------------------------------------
extern "C" void kernel_launch(void* const* d_in, const int* in_sizes, int n_in,
                              void* d_out, int out_size, void* d_ws, size_t ws_size,
                              hipStream_t stream)
{
    const float* x   = (const float*)d_in[0];
    const float* rpn = (const float*)d_in[1];
    const float* w1  = (const float*)d_in[2];
    const float* b1  = (const float*)d_in[3];
    const float* w2  = (const float*)d_in[4];
    const float* b2  = (const float*)d_in[5];
    const float* w3  = (const float*)d_in[6];
    const float* b3  = (const float*)d_in[7];
    const float* tw1 = (const float*)d_in[8];
    const float* tb1 = (const float*)d_in[9];
    const float* tw2 = (const float*)d_in[10];
    const float* tb2 = (const float*)d_in[11];
    const float* tw3 = (const float*)d_in[12];
    const float* tb3 = (const float*)d_in[13];
    const int*   ea  = (const int*)d_in[14];

    // workspace layout (floats):
    //   d1 | d2 | d3 | scores | topIdx(pad) | wt1 | wt2 | wt3
    float* d1 = (float*)d_ws;               // 8*128*196 = 200704
    float* d2 = d1 + 200704;                // 200704
    float* d3 = d2 + 200704;                // 8*128*49 = 50176
    float* sc = d3 + 50176;                 // 8*2793 (padded to 22400)
    int*   ti = (int*)(sc + 22400);         // 32 ints (padded to 64 floats)
    float* wt1 = (float*)(sc + 22400 + 64); // 9*128*2048 = 2359296
    float* wt2 = wt1 + 2359296;             // 9*128*128  = 147456
    float* wt3 = wt2 + 147456;              // 147456

    const size_t needPacked = (size_t)(474112 + 2359296 + 2 * 147456) * sizeof(float);
    const bool packed = (ws_size >= needPacked);

    float* out = (float*)d_out;
    const int P0 = BATCH * 4 * 3 * OUTS * OUTS;   // part_imgs element count

    if (packed) {
        // one-time (per launch) weight transposes: w[o][c][r] -> wt[r][o][c]
        repack_w<<<(9 * 128 * 2048 + 255) / 256, 256, 0, stream>>>(w1, wt1, 2048);
        repack_w<<<(9 * 128 *  128 + 255) / 256, 256, 0, stream>>>(w2, wt2, 128);
        repack_w<<<(9 * 128 *  128 + 255) / 256, 256, 0, stream>>>(w3, wt3, 128);

        conv3x3_relu_wmma<true><<<dim3(13, BATCH), 256, 0, stream>>>(rpn, wt1, b1, d1, 2048, 14, 14, 1);
        conv3x3_relu_wmma<true><<<dim3(13, BATCH), 256, 0, stream>>>(d1,  wt2, b2, d2,  128, 14, 14, 1);
        conv3x3_relu_wmma<true><<<dim3( 4, BATCH), 256, 0, stream>>>(d2,  wt3, b3, d3,  128, 14,  7, 2);
    } else {
        conv3x3_relu_wmma<false><<<dim3(13, BATCH), 256, 0, stream>>>(rpn, w1, b1, d1, 2048, 14, 14, 1);
        conv3x3_relu_wmma<false><<<dim3(13, BATCH), 256, 0, stream>>>(d1,  w2, b2, d2,  128, 14, 14, 1);
        conv3x3_relu_wmma<false><<<dim3( 4, BATCH), 256, 0, stream>>>(d2,  w3, b3, d3,  128, 14,  7, 2);
    }

    // score heads -> concat(t1,t2,t3)
    scores_kernel<<<(BATCH * NANCH + 255) / 256, 256, 0, stream>>>(
        d1, d2, d3, tw1, tb1, tw2, tb2, tw3, tb3, sc);

    // top-4 NMS per image; writes top_n_prob / top_n_index tail of d_out
    nms_kernel<<<BATCH, 256, 0, stream>>>(sc, ea, ti, out + P0);

    // bilinear crop-resize of the 32 selected boxes
    crop_kernel<<<(32 * 3 * OUTS * OUTS) / 256, 256, 0, stream>>>(x, ea, ti, out);
}